// U_CFM_57784490000583
// MI455X (gfx1250) — compile-verified
//
#include <hip/hip_runtime.h>
#include <hip/hip_bf16.h>

typedef float v2f __attribute__((ext_vector_type(2)));
typedef float v4f __attribute__((ext_vector_type(4)));
typedef float v8f __attribute__((ext_vector_type(8)));

// U_CFM fused: x = x_enc + ctx; per (b,c) row of length T=16:
// 4x [Linear -> LayerNorm -> ReLU] with dims 16->11->6->11->16.
//
// One wave32 owns a 32-row super-tile (= two 16x16 WMMA A-tiles), looping over
// KT super-tiles so the masked W^T fragments (hoisted in VGPRs) amortize.
// Each Linear = 4 chained V_WMMA_F32_16X16X4_F32 per tile. D tiles are
// transposed through a per-wave LDS slab; every lane then owns a DISTINCT row,
// so LayerNorm is non-redundant straight-line VALU with scalar-loaded params.
__global__ __launch_bounds__(256) void ucfm_fused(
    const float* __restrict__ x, const float* __restrict__ ctx,
    const float* __restrict__ W0, const float* __restrict__ b0,
    const float* __restrict__ g0, const float* __restrict__ e0,
    const float* __restrict__ W1, const float* __restrict__ b1,
    const float* __restrict__ g1, const float* __restrict__ e1,
    const float* __restrict__ W2, const float* __restrict__ b2,
    const float* __restrict__ g2, const float* __restrict__ e2,
    const float* __restrict__ W3, const float* __restrict__ b3,
    const float* __restrict__ g3, const float* __restrict__ e3,
    float* __restrict__ out, int kt)
{
    constexpr int T = 16;
    constexpr int C = 2048;
    constexpr int RS = 20;   // LDS row stride (floats): 16B-aligned
    __shared__ __align__(16) float lds[8][32 * RS];

    const int lane = threadIdx.x & 31;
    const int wave = threadIdx.x >> 5;
    const int m = lane & 15;                  // N column / A row within a tile
    const int h = lane >> 4;                  // wave half (K split for A/B)

    float* slab = lds[wave];

    const int inf[4]  = {16, 11, 6, 11};
    const int outf[4] = {11, 6, 11, 16};
    const float* const Ws[4] = {W0, W1, W2, W3};
    const float* const Bi[4] = {b0, b1, b2, b3};
    const float* const Ga[4] = {g0, g1, g2, g3};
    const float* const Be[4] = {e0, e1, e2, e3};

    // ---- Hoisted B fragments: W^T zero-padded to 16x16, per layer.
    // Clamped index + float mask keeps loads in-bounds, EXEC all-ones.
    v2f bf[4][4];
#pragma unroll
    for (int l = 0; l < 4; ++l) {
        const int in_f = inf[l], out_f = outf[l];
        const int n = m;
#pragma unroll
        for (int c = 0; c < 4; ++c) {
            const int k0 = 4 * c + 2 * h;
            const int nc  = (n < out_f) ? n : (out_f - 1);
            const int kc0 = (k0 < in_f) ? k0 : (in_f - 1);
            const int kc1 = (k0 + 1 < in_f) ? (k0 + 1) : (in_f - 1);
            const float m0 = (k0 < in_f && n < out_f) ? 1.0f : 0.0f;
            const float m1 = (k0 + 1 < in_f && n < out_f) ? 1.0f : 0.0f;
            v2f w;
            w.x = Ws[l][nc * in_f + kc0] * m0;
            w.y = Ws[l][nc * in_f + kc1] * m1;
            bf[l][c] = w;
        }
    }

    const long gw = (long)blockIdx.x * 8 + wave;   // global wave id

#pragma unroll 1
    for (int it = 0; it < kt; ++it) {
        const size_t r0 = ((size_t)gw * kt + it) * 32;    // 32 rows/super-tile
        const size_t base = (r0 >> 11) * (T * C) + (r0 & (C - 1));

        // ---- Input: dense 128-bit loads (4x128B per instr) + LDS transpose.
        // Lane (cg = lane&7, tg = lane>>3) loads a 16B c-chunk at rows t=tg+4j.
        {
            const int cg = lane & 7;
            const int tg = lane >> 3;
#pragma unroll
            for (int j = 0; j < 4; ++j) {
                const int t = tg + 4 * j;
                const size_t p = base + (size_t)t * C + (size_t)(4 * cg);
                const v4f s = *(const v4f*)(x + p) + *(const v4f*)(ctx + p);
#pragma unroll
                for (int i = 0; i < 4; ++i)
                    slab[(4 * cg + i) * RS + t] = s[i];
            }
        }

        float row[16];
#pragma unroll
        for (int l = 0; l < 4; ++l) {
            const int out_f = outf[l];

            // A fragments straight from the slab (8B, conflict-free), two
            // tiles: rows 0-15 and rows 16-31. Same B fragment reused.
            v8f d0 = {}, d1 = {};
#pragma unroll
            for (int c = 0; c < 4; ++c) {
                const int ko = 4 * c + 2 * h;
                const v2f a0 = *(const v2f*)(slab + m * RS + ko);
                const v2f a1 = *(const v2f*)(slab + (16 + m) * RS + ko);
                d0 = __builtin_amdgcn_wmma_f32_16x16x4_f32(
                        false, a0, false, bf[l][c], (short)0, d0, false, false);
                d1 = __builtin_amdgcn_wmma_f32_16x16x4_f32(
                        false, a1, false, bf[l][c], (short)0, d1, false, false);
            }

            // Scatter both D tiles: row (tile*16 + r + 8h), col n=m.
#pragma unroll
            for (int r = 0; r < 8; ++r) {
                slab[(r + 8 * h) * RS + m]        = d0[r];
                slab[(16 + r + 8 * h) * RS + m]   = d1[r];
            }

            // Each lane gathers its OWN row (lane = row 0..31): LN is
            // computed exactly once per row across the wave.
#pragma unroll
            for (int j = 0; j < 4; ++j) {
                const v4f r4 = *(const v4f*)(slab + lane * RS + 4 * j);
                row[4 * j + 0] = r4.x; row[4 * j + 1] = r4.y;
                row[4 * j + 2] = r4.z; row[4 * j + 3] = r4.w;
            }

            // Bias + LayerNorm + ReLU, in-lane. i/out_f are compile-time, so
            // param reads become uniform scalar loads; padded features fold.
            float s = 0.0f;
#pragma unroll
            for (int i = 0; i < 16; ++i) {
                if (i < out_f) { row[i] += Bi[l][i]; s += row[i]; }
                else           { row[i] = 0.0f; }
            }
            const float rinv = 1.0f / (float)out_f;
            const float mu = s * rinv;
            float q = 0.0f;
#pragma unroll
            for (int i = 0; i < 16; ++i)
                if (i < out_f) q = fmaf(row[i], row[i], q);
            const float rstd = rsqrtf(q * rinv - mu * mu + 1e-5f);
#pragma unroll
            for (int i = 0; i < 16; ++i) {
                if (i < out_f) {
                    const float z = (row[i] - mu) * rstd * Ga[l][i] + Be[l][i];
                    row[i] = fmaxf(z, 0.0f);
                }
            }

            if (l < 3) {
                // Write the normalized row back for next layer's A gathers.
#pragma unroll
                for (int j = 0; j < 4; ++j) {
                    v4f r4;
                    r4.x = row[4 * j + 0]; r4.y = row[4 * j + 1];
                    r4.z = row[4 * j + 2]; r4.w = row[4 * j + 3];
                    *(v4f*)(slab + lane * RS + 4 * j) = r4;
                }
            }
        }

        // ---- Output straight from row registers: 32 contiguous c at fixed t
        // = one dense, 128B-aligned line per store instruction.
        float* ob = out + base;
#pragma unroll
        for (int t = 0; t < 16; ++t)
            ob[(size_t)t * C + (size_t)lane] = row[t];
    }
}

extern "C" void kernel_launch(void* const* d_in, const int* in_sizes, int n_in,
                              void* d_out, int out_size, void* d_ws, size_t ws_size,
                              hipStream_t stream) {
    const float* x   = (const float*)d_in[0];
    const float* ctx = (const float*)d_in[1];
    const float* W0 = (const float*)d_in[2];
    const float* b0 = (const float*)d_in[3];
    const float* g0 = (const float*)d_in[4];
    const float* e0 = (const float*)d_in[5];
    const float* W1 = (const float*)d_in[6];
    const float* b1 = (const float*)d_in[7];
    const float* g1 = (const float*)d_in[8];
    const float* e1 = (const float*)d_in[9];
    const float* W2 = (const float*)d_in[10];
    const float* b2 = (const float*)d_in[11];
    const float* g2 = (const float*)d_in[12];
    const float* e2 = (const float*)d_in[13];
    const float* W3 = (const float*)d_in[14];
    const float* b3 = (const float*)d_in[15];
    const float* g3 = (const float*)d_in[16];
    const float* e3 = (const float*)d_in[17];
    float* out = (float*)d_out;

    // total = B*T*C; rows = total/16; super-tiles = rows/32; KT per wave;
    // 8 waves per block -> blocks = total / (16*32*KT*8).
    constexpr int KT = 4;
    const long total = (long)in_sizes[0];
    const int blocks = (int)(total / (16L * 32L * KT * 8L));
    ucfm_fused<<<blocks, 256, 0, stream>>>(x, ctx,
                                           W0, b0, g0, e0,
                                           W1, b1, g1, e1,
                                           W2, b2, g2, e2,
                                           W3, b3, g3, e3,
                                           out, KT);
}